// CIN_11510512353904
// MI455X (gfx1250) — compile-verified
//
#include <hip/hip_runtime.h>

typedef float v2f __attribute__((ext_vector_type(2)));
typedef float v4f __attribute__((ext_vector_type(4)));
typedef float v8f __attribute__((ext_vector_type(8)));

#define CIN_B    2048
#define CIN_F0   39
#define CIN_D    16
#define CIN_M    128
#define NB       4          // batches per wave (W-fragment reuse factor)
#define NTHREADS 256        // 8 waves -> 8 m-tiles of 16 = M=128
#define XKT_STRIDE (CIN_M + 4)   // padded [d][g] stride (dwords) -> conflict-free b128

// One CIN layer as a fused outer-product GEMM:
//   cur[b,m,d] = sum_{f<F0, g<G} x0[b,f,d] * xk[b,g,d] * W[f*G+g, m]
// GEMM rows are (b,d) pairs; 16 rows == one batch element (d=0..15).
// Each wave computes a 16(d) x 16(m) fp32 WMMA tile for NB batches.
template <int G>
__launch_bounds__(NTHREADS)
__global__ void cin_layer_kernel(const float* __restrict__ x0,
                                 const float* __restrict__ xk,
                                 const float* __restrict__ W,
                                 float* __restrict__ cur_out,   // [B,128,16] or nullptr
                                 float* __restrict__ out,       // [B,384]
                                 int layer_off)
{
    constexpr int P = CIN_F0 * G;          // K extent of the GEMM
    constexpr bool BIG = (G == CIN_M);     // layers 2/3: structured f/g loops

    __shared__ float x0s[NB * CIN_F0 * CIN_D];                       // [nb][f][d]
    __shared__ float xks[BIG ? (NB * CIN_D * XKT_STRIDE)             // [nb][d][g] (transposed)
                             : (NB * G * CIN_D)];                    // [nb][g][d]

    const int bBase = blockIdx.x * NB;
    const int tid   = threadIdx.x;

    // ---- stage tiles for NB consecutive batches ----
    {
        const float* g0 = x0 + (size_t)bBase * CIN_F0 * CIN_D;
        for (int i = tid; i < NB * CIN_F0 * CIN_D; i += NTHREADS) x0s[i] = g0[i];
        const float* g1 = xk + (size_t)bBase * G * CIN_D;
        if constexpr (BIG) {
            // transpose: global [nb][g][d] -> LDS [nb][d][g] (padded)
            for (int i = tid; i < NB * G * CIN_D; i += NTHREADS) {
                int nb = i >> 11;              // /(128*16)
                int rem = i & 2047;
                int g = rem >> 4, dd = rem & 15;
                xks[nb * CIN_D * XKT_STRIDE + dd * XKT_STRIDE + g] = g1[i];
            }
        } else {
            for (int i = tid; i < NB * G * CIN_D; i += NTHREADS) xks[i] = g1[i];
        }
    }
    __syncthreads();

    const int wave = tid >> 5;             // 0..7 -> m-tile
    const int lane = tid & 31;
    const int m0   = wave * 16;
    const int d    = lane & 15;            // A/C row (embedding dim); also W column lane
    const int kSel = lane >> 4;            // fragment half: positions {0,1} vs {2,3}

    v8f acc[NB];
    {
        v8f z = {};
        #pragma unroll
        for (int nb = 0; nb < NB; ++nb) acc[nb] = z;
    }

    const float* Wc = W + m0 + d;          // this lane's W column

    if constexpr (BIG) {
        // ---- layers 2/3: G=128. K-chunk pair jj covers g = {4jj..4jj+3} (+64 for
        // kSel=1 lanes), so each lane reads 4 contiguous g -> one ds_load_b128
        // feeding two WMMAs. A/B sides use the same position<->K mapping. ----
        #pragma unroll 1
        for (int f = 0; f < CIN_F0; ++f) {
            float x0v[NB];
            #pragma unroll
            for (int nb = 0; nb < NB; ++nb)
                x0v[nb] = x0s[nb * CIN_F0 * CIN_D + f * CIN_D + d];
            const float* Wf = Wc + (size_t)f * G * CIN_M;
            #pragma unroll 2
            for (int jj = 0; jj < 16; ++jj) {
                const int gbase = 4 * jj + 64 * kSel;   // lane's 4 contiguous g
                v2f bf0, bf1;
                bf0.x = Wf[(size_t)(gbase + 0) * CIN_M];
                bf0.y = Wf[(size_t)(gbase + 1) * CIN_M];
                bf1.x = Wf[(size_t)(gbase + 2) * CIN_M];
                bf1.y = Wf[(size_t)(gbase + 3) * CIN_M];
                #pragma unroll
                for (int nb = 0; nb < NB; ++nb) {
                    const v4f xv = *reinterpret_cast<const v4f*>(
                        &xks[nb * CIN_D * XKT_STRIDE + d * XKT_STRIDE + gbase]);
                    const v2f af0 = (v2f){xv.x, xv.y} * x0v[nb];   // v_pk_mul_f32
                    acc[nb] = __builtin_amdgcn_wmma_f32_16x16x4_f32(
                        false, af0, false, bf0, (short)0, acc[nb], false, false);
                    const v2f af1 = (v2f){xv.z, xv.w} * x0v[nb];
                    acc[nb] = __builtin_amdgcn_wmma_f32_16x16x4_f32(
                        false, af1, false, bf1, (short)0, acc[nb], false, false);
                }
            }
        }
    } else {
        // ---- layer 1: G=39, flat K loop with incremental (f,g), no division ----
        constexpr int PMAIN = P & ~3;      // 1520
        int f0 = 0, g0 = 2 * kSel;         // tracks r0 = p + 2*kSel
        #pragma unroll 2
        for (int p = 0; p < PMAIN; p += 4) {
            const int r0 = p + 2 * kSel;
            // (f1,g1) for r0+1 without division (g0 <= G-1 -> g0+1 <= G)
            const bool c1 = (g0 + 1 == G);
            const int g1 = c1 ? 0 : g0 + 1;
            const int f1 = c1 ? f0 + 1 : f0;

            v2f bf;
            bf.x = Wc[(size_t)r0 * CIN_M];
            bf.y = Wc[(size_t)(r0 + 1) * CIN_M];
            #pragma unroll
            for (int nb = 0; nb < NB; ++nb) {
                const v2f xa = { x0s[nb * CIN_F0 * CIN_D + f0 * CIN_D + d],
                                 x0s[nb * CIN_F0 * CIN_D + f1 * CIN_D + d] };
                const v2f xb = { xks[nb * G * CIN_D + g0 * CIN_D + d],
                                 xks[nb * G * CIN_D + g1 * CIN_D + d] };
                const v2f af = xa * xb;                 // v_pk_mul_f32
                acc[nb] = __builtin_amdgcn_wmma_f32_16x16x4_f32(
                    false, af, false, bf, (short)0, acc[nb], false, false);
            }
            // advance r0 by 4: g0 += 4 with one wrap (step < G)
            g0 += 4;
            const bool c = (g0 >= G);
            g0 = c ? g0 - G : g0;
            f0 = c ? f0 + 1 : f0;
        }
        if constexpr ((P & 3) != 0) {
            // tail: single valid K index PMAIN (f=g=F0-1); mask A side, no branches
            const float msk = (kSel == 0) ? 1.0f : 0.0f;   // only .x of kSel==0 valid
            v2f bf;
            bf.x = Wc[(size_t)PMAIN * CIN_M];               // same row for both halves
            bf.y = 0.0f;
            constexpr int ft = PMAIN / G, gt = PMAIN % G;   // 38, 38
            #pragma unroll
            for (int nb = 0; nb < NB; ++nb) {
                v2f af;
                af.x = msk * x0s[nb * CIN_F0 * CIN_D + ft * CIN_D + d] *
                             xks[nb * G      * CIN_D + gt * CIN_D + d];
                af.y = 0.0f;
                acc[nb] = __builtin_amdgcn_wmma_f32_16x16x4_f32(
                    false, af, false, bf, (short)0, acc[nb], false, false);
            }
        }
    }

    // ---- epilogue: C/D layout -> VGPR i holds (d = i + 8*kSel, m = m0 + (lane&15)) ----
    #pragma unroll
    for (int nb = 0; nb < NB; ++nb) {
        const int b = bBase + nb;
        const int m = m0 + d;

        if (cur_out != nullptr) {
            float* cp = cur_out + ((size_t)b * CIN_M + m) * CIN_D + kSel * 8;
            float4 lo = make_float4(acc[nb][0], acc[nb][1], acc[nb][2], acc[nb][3]);
            float4 hi = make_float4(acc[nb][4], acc[nb][5], acc[nb][6], acc[nb][7]);
            ((float4*)cp)[0] = lo;
            ((float4*)cp)[1] = hi;
        }

        // out[b, layer_off + m] = sum_d cur[b,m,d]
        v2f t = (v2f){acc[nb][0], acc[nb][1]} + (v2f){acc[nb][2], acc[nb][3]};
        t    += (v2f){acc[nb][4], acc[nb][5]} + (v2f){acc[nb][6], acc[nb][7]};
        float s = t.x + t.y;
        s += __shfl_xor(s, 16, 32);        // combine the two d-halves (wave32)
        if (lane < 16) {
            out[(size_t)b * 384 + layer_off + m0 + lane] = s;
        }
    }
}

extern "C" void kernel_launch(void* const* d_in, const int* in_sizes, int n_in,
                              void* d_out, int out_size, void* d_ws, size_t ws_size,
                              hipStream_t stream) {
    const float* x  = (const float*)d_in[0];   // [2048, 39, 16]
    const float* f0 = (const float*)d_in[1];   // [1521, 128]
    const float* f1 = (const float*)d_in[2];   // [4992, 128]
    const float* f2 = (const float*)d_in[3];   // [4992, 128]
    float* out = (float*)d_out;                // [2048, 384]

    float* cur1 = (float*)d_ws;                         // [2048,128,16] fp32
    float* cur2 = cur1 + (size_t)CIN_B * CIN_M * CIN_D; // [2048,128,16] fp32

    dim3 grid(CIN_B / NB);
    dim3 block(NTHREADS);

    cin_layer_kernel<CIN_F0><<<grid, block, 0, stream>>>(x, x,    f0, cur1,    out, 0);
    cin_layer_kernel<CIN_M ><<<grid, block, 0, stream>>>(x, cur1, f1, cur2,    out, 128);
    cin_layer_kernel<CIN_M ><<<grid, block, 0, stream>>>(x, cur2, f2, nullptr, out, 256);
}